// Camera_36670430773890
// MI455X (gfx1250) — compile-verified
//
#include <hip/hip_runtime.h>
#include <cstdint>
#include <cstddef>

#define TILE 256

// Gaussian-splat camera projection: pos2d (N,3), cov2d (N,2,2), mask (N) from
// pos3d (N,3), cov3d (N,3,3), view (4,4), proj (4,4).
//
// Bandwidth-bound streaming kernel (80 B/point -> 320 MB/launch -> ~13.7 us
// floor at 23.3 TB/s). Both input and output traffic go through the gfx1250
// async LDS DMA path: global_load_async_to_lds_b128 for the strided inputs,
// global_store_async_from_lds_b128 for the outputs, so every global
// transaction is 16 B/lane over contiguous tile regions.
__global__ __launch_bounds__(TILE)
void camera_proj_kernel(const float* __restrict__ pos3d,
                        const float* __restrict__ cov3d,
                        const float* __restrict__ viewm,
                        const float* __restrict__ projm,
                        float* __restrict__ pos2d,
                        float* __restrict__ cov2d,
                        float* __restrict__ maskf,
                        int N)
{
    __shared__ __align__(16) float sPos[TILE * 3];  // 3 KB  (input stage)
    __shared__ __align__(16) float sCov[TILE * 9];  // 9 KB  (input stage)
    __shared__ __align__(16) float sP2 [TILE * 3];  // 3 KB  (output stage)
    __shared__ __align__(16) float sC2 [TILE * 4];  // 4 KB  (output stage)
    __shared__ __align__(16) float sM  [TILE];      // 1 KB  (output stage)

    const int tid = threadIdx.x;
    const long long tile0 = (long long)blockIdx.x * TILE;
    const bool full_tile = (tile0 + TILE) <= (long long)N;
    // Async/vector output path needs 16B alignment of all three output
    // sub-arrays: holds iff N % 4 == 0 (true for the harness N = 4,000,000).
    const bool vec_out = full_tile && ((N & 3) == 0);

    // Uniform matrices: uniform-address loads -> scalar (s_load) path.
    float V[16], P[16];
#pragma unroll
    for (int i = 0; i < 16; ++i) { V[i] = viewm[i]; P[i] = projm[i]; }

    if (full_tile) {
        // ---- async DMA staging: memory -> LDS, 16B per lane, contiguous ----
        const float* gp = pos3d + tile0 * 3;   // 192 float4 (3072B tile, 16B-aligned)
        const float* gc = cov3d + tile0 * 9;   // 576 float4 (9216B tile, 16B-aligned)
        if (tid < (TILE * 3) / 4) {
            uint32_t lds_a = (uint32_t)(uintptr_t)(&sPos[tid * 4]);
            uint64_t ga    = (uint64_t)(uintptr_t)(gp + tid * 4);
            asm volatile("global_load_async_to_lds_b128 %0, %1, off"
                         :: "v"(lds_a), "v"(ga) : "memory");
        }
#pragma unroll
        for (int k = 0; k < 3; ++k) {
            int i = tid + k * TILE;
            if (i < (TILE * 9) / 4) {
                uint32_t lds_a = (uint32_t)(uintptr_t)(&sCov[i * 4]);
                uint64_t ga    = (uint64_t)(uintptr_t)(gc + i * 4);
                asm volatile("global_load_async_to_lds_b128 %0, %1, off"
                             :: "v"(lds_a), "v"(ga) : "memory");
            }
        }
        asm volatile("s_wait_asynccnt 0" ::: "memory");
    } else {
        // Tail tile (never taken for N % TILE == 0): guarded direct loads.
        int n = (int)tile0 + tid;
        if (n < N) {
            const float* gp = pos3d + (long long)n * 3;
            const float* gc = cov3d + (long long)n * 9;
#pragma unroll
            for (int k = 0; k < 3; ++k) sPos[tid * 3 + k] = gp[k];
#pragma unroll
            for (int k = 0; k < 9; ++k) sCov[tid * 9 + k] = gc[k];
        }
    }
    __syncthreads();

    const int n = (int)(tile0 + tid);
    const bool valid = (n < N);

    // LDS reads: strides 3 and 9 words are coprime with banks -> conflict-free.
    const float x = sPos[tid * 3 + 0];
    const float y = sPos[tid * 3 + 1];
    const float z = sPos[tid * 3 + 2];

    // p_model = view[:3,:3] @ pos + view[:3,3]
    const float pv0 = V[0] * x + V[1] * y + V[2]  * z + V[3];
    const float pv1 = V[4] * x + V[5] * y + V[6]  * z + V[7];
    const float pv2 = V[8] * x + V[9] * y + V[10] * z + V[11];

    // p_clip = proj @ (p_model, 1)
    const float c0 = P[0]  * pv0 + P[1]  * pv1 + P[2]  * pv2 + P[3];
    const float c1 = P[4]  * pv0 + P[5]  * pv1 + P[6]  * pv2 + P[7];
    const float c2 = P[8]  * pv0 + P[9]  * pv1 + P[10] * pv2 + P[11];
    const float c3 = P[12] * pv0 + P[13] * pv1 + P[14] * pv2 + P[15];

    const float invw = 1.0f / (c3 + 1e-6f);
    const float ndx = c0 * invw;
    const float ndy = c1 * invw;
    const float ndz = c2 * invw;

    const bool m = (ndz >= 0.2f) &&
                   (ndx >= -1.3f) && (ndx <= 1.3f) &&
                   (ndy >= -1.3f) && (ndy <= 1.3f);

    const float sx = 0.5f * (ndx + 1.0f) * 1920.0f;
    const float sy = (1.0f - 0.5f * (ndy + 1.0f)) * 1080.0f;

    // cov2d = (J @ Wm) C (J @ Wm)^T,  Wm = view[:3,:3].T  (Wm[j][c] = V[c*4+j])
    const float iz = 1.0f / z;
    const float a = -x * iz;   // J[0] = (iz, 0, a)
    const float b = -y * iz;   // J[1] = (0, iz, b)

    const float u00 = iz * V[0] + a * V[2];
    const float u01 = iz * V[4] + a * V[6];
    const float u02 = iz * V[8] + a * V[10];
    const float u10 = iz * V[1] + b * V[2];
    const float u11 = iz * V[5] + b * V[6];
    const float u12 = iz * V[9] + b * V[10];

    const float* C = &sCov[tid * 9];  // row-major 3x3 (symmetric)
    const float t00 = u00 * C[0] + u01 * C[3] + u02 * C[6];
    const float t01 = u00 * C[1] + u01 * C[4] + u02 * C[7];
    const float t02 = u00 * C[2] + u01 * C[5] + u02 * C[8];
    const float t10 = u10 * C[0] + u11 * C[3] + u12 * C[6];
    const float t11 = u10 * C[1] + u11 * C[4] + u12 * C[7];
    const float t12 = u10 * C[2] + u11 * C[5] + u12 * C[8];

    const float o00 = t00 * u00 + t01 * u01 + t02 * u02;
    const float o01 = t00 * u10 + t01 * u11 + t02 * u12;
    const float o10 = t10 * u00 + t11 * u01 + t12 * u02;
    const float o11 = t10 * u10 + t11 * u11 + t12 * u12;

    const float p0 = m ? sx  : 0.0f;
    const float p1 = m ? sy  : 0.0f;
    const float p2 = m ? ndz : 0.0f;
    float4 cv;
    if (m) { cv.x = o00; cv.y = o01; cv.z = o10; cv.w = o11; }
    else   { cv.x = 0.0f; cv.y = 0.0f; cv.z = 0.0f; cv.w = 0.0f; }
    const float fm = m ? 1.0f : 0.0f;

    if (vec_out) {
        // ---- stage outputs in LDS, then async DMA: LDS -> memory ----
        sP2[tid * 3 + 0] = p0;
        sP2[tid * 3 + 1] = p1;
        sP2[tid * 3 + 2] = p2;
        reinterpret_cast<float4*>(sC2)[tid] = cv;   // ds_store_b128
        sM[tid] = fm;
        __syncthreads();   // all waves' results visible in LDS

        if (tid < (TILE * 3) / 4) {                  // 192 float4
            uint64_t ga    = (uint64_t)(uintptr_t)(pos2d + tile0 * 3 + tid * 4);
            uint32_t lds_a = (uint32_t)(uintptr_t)(&sP2[tid * 4]);
            asm volatile("global_store_async_from_lds_b128 %0, %1, off"
                         :: "v"(ga), "v"(lds_a) : "memory");
        }
        {                                            // 256 float4
            uint64_t ga    = (uint64_t)(uintptr_t)(cov2d + tile0 * 4 + tid * 4);
            uint32_t lds_a = (uint32_t)(uintptr_t)(&sC2[tid * 4]);
            asm volatile("global_store_async_from_lds_b128 %0, %1, off"
                         :: "v"(ga), "v"(lds_a) : "memory");
        }
        if (tid < TILE / 4) {                        // 64 float4
            uint64_t ga    = (uint64_t)(uintptr_t)(maskf + tile0 + tid * 4);
            uint32_t lds_a = (uint32_t)(uintptr_t)(&sM[tid * 4]);
            asm volatile("global_store_async_from_lds_b128 %0, %1, off"
                         :: "v"(ga), "v"(lds_a) : "memory");
        }
        // ASYNCcnt drained by the implicit wait-idle at s_endpgm.
    } else if (valid) {
        // Scalar fallback (tail tiles / unaligned N).
        const size_t bn = (size_t)n;
        pos2d[bn * 3 + 0] = p0;
        pos2d[bn * 3 + 1] = p1;
        pos2d[bn * 3 + 2] = p2;
        cov2d[bn * 4 + 0] = cv.x;
        cov2d[bn * 4 + 1] = cv.y;
        cov2d[bn * 4 + 2] = cv.z;
        cov2d[bn * 4 + 3] = cv.w;
        maskf[bn] = fm;
    }
}

extern "C" void kernel_launch(void* const* d_in, const int* in_sizes, int n_in,
                              void* d_out, int out_size, void* d_ws, size_t ws_size,
                              hipStream_t stream)
{
    const float* pos3d = (const float*)d_in[0];   // (N,3)
    const float* cov3d = (const float*)d_in[1];   // (N,3,3)
    const float* viewm = (const float*)d_in[2];   // (4,4)
    const float* projm = (const float*)d_in[3];   // (4,4)

    const int N = in_sizes[0] / 3;

    float* out   = (float*)d_out;
    float* pos2d = out;                      // N*3
    float* cov2d = out + (size_t)3 * N;      // N*4
    float* maskf = out + (size_t)7 * N;      // N

    const int grid = (N + TILE - 1) / TILE;
    camera_proj_kernel<<<grid, TILE, 0, stream>>>(pos3d, cov3d, viewm, projm,
                                                  pos2d, cov2d, maskf, N);
}